// I_MLP_7206955123212
// MI455X (gfx1250) — compile-verified
//
#include <hip/hip_runtime.h>

// ---------------------------------------------------------------------------
// CDNA5 (gfx1250) rollout kernel: 10-step tiny-MLP dynamics per batch element.
// Weights live in WMMA A-matrices (M = out features, K = in features, padded
// to 16x4); 16 batch elements ride as the N columns of B/D. Every layer GEMM
// is a v_wmma_f32_16x16x4_f32; lane-half swaps (ds_swizzle SWAPX16, LDS pipe)
// rebuild the next layer's B operand from the previous D result; activations
// use native v_tanh_f32. fc1 and hf1 share the same B operand, so their
// weight matrices are stacked into one A (rows 0-3 / 4-7) -> 8 WMMAs/step.
// One wave = 16 rollouts; all weights/biases register/SGPR-resident.
// ---------------------------------------------------------------------------

typedef float v2f __attribute__((ext_vector_type(2)));
typedef float v8f __attribute__((ext_vector_type(8)));

#define ROLL_STEPS 10

__device__ __forceinline__ float tanh_f(float x) {
#if __has_builtin(__builtin_amdgcn_tanhf)
  return __builtin_amdgcn_tanhf(x);            // v_tanh_f32 (gfx1250 trans op)
#else
  float ax = __builtin_fabsf(x);
  float t  = __builtin_amdgcn_exp2f(-2.8853900817779268f * ax); // e^(-2|x|)
  float y  = (1.0f - t) * __builtin_amdgcn_rcpf(1.0f + t);
  return __builtin_copysignf(y, x);
#endif
}

// swap lane halves (lane ^= 16): group-of-32 ds_swizzle, xor=0x10, and=0x1f
__device__ __forceinline__ float swap16f(float x) {
  return __int_as_float(__builtin_amdgcn_ds_swizzle(__float_as_int(x), 0x401F));
}

// Build one B-matrix VGPR: lanes 0-15 take 'lo' (K row k), lanes 16-31 take
// 'hisrc' moved up from lanes 0-15 (K row k+2). Swizzle runs at full EXEC.
__device__ __forceinline__ float upmix(float lo, float hisrc, bool hi) {
  float sw = swap16f(hisrc);
  return hi ? sw : lo;
}

__device__ __forceinline__ v8f wmma4(v2f a, v2f b, v8f c) {
  // (neg_a, A, neg_b, B, c_mod, C, reuse_a, reuse_b)
  return __builtin_amdgcn_wmma_f32_16x16x4_f32(false, a, false, b, (short)0, c,
                                               false, false);
}

// Load rows of W (row-major [rows x cols]) into A-matrix 16x4 f32 layout:
// lanes 0-15: V0=K(koff), V1=K(koff+1); lanes 16-31: V0=K(koff+2), V1=K(koff+3).
// Rows >= 'rows' are zero-padded.
__device__ __forceinline__ v2f load_A(const float* W, int rows, int cols,
                                      int koff, int lane) {
  int r = lane & 15;
  int k = koff + ((lane >> 4) << 1);
  bool ok = r < rows;
  v2f a;
  a[0] = ok ? W[r * cols + k + 0] : 0.0f;
  a[1] = ok ? W[r * cols + k + 1] : 0.0f;
  return a;
}

// Two 4-row matrices (same K=4) stacked into rows 0-3 / 4-7 of one A operand.
__device__ __forceinline__ v2f load_A_stacked(const float* Wlo,
                                              const float* Whi, int lane) {
  int r = lane & 15;
  int k = (lane >> 4) << 1;
  v2f a;
  const float* W = (r < 4) ? (Wlo + r * 4) : (Whi + (r - 4) * 4);
  bool ok = r < 8;
  a[0] = ok ? W[k + 0] : 0.0f;
  a[1] = ok ? W[k + 1] : 0.0f;
  return a;
}

// Bias vector in C/D layout: VGPR r broadcasts b[r] (feature r, lanes 0-15).
template <int N>
__device__ __forceinline__ v8f bias_C(const float* b) {
  v8f c;
#pragma unroll
  for (int r = 0; r < 8; ++r) c[r] = (r < N) ? b[r] : 0.0f;
  return c;
}

// Stacked bias: rows 0-3 from b_lo (4), rows 4-7 from b_hi (4).
__device__ __forceinline__ v8f bias_C_stacked(const float* b_lo,
                                              const float* b_hi) {
  v8f c;
#pragma unroll
  for (int r = 0; r < 4; ++r) c[r] = b_lo[r];
#pragma unroll
  for (int r = 0; r < 4; ++r) c[4 + r] = b_hi[r];
  return c;
}

__global__ __launch_bounds__(256) void rollout_wmma_kernel(
    const float* __restrict__ s_star, const float* __restrict__ s0,
    const float* __restrict__ fc1_w, const float* __restrict__ fc1_b,
    const float* __restrict__ fc2_w, const float* __restrict__ fc2_b,
    const float* __restrict__ rc1_w, const float* __restrict__ rc1_b,
    const float* __restrict__ rc2_w, const float* __restrict__ rc2_b,
    const float* __restrict__ rc3_w, const float* __restrict__ rc3_b,
    const float* __restrict__ hf1_w, const float* __restrict__ hf1_b,
    const float* __restrict__ hf2_w, const float* __restrict__ hf2_b,
    float* __restrict__ err_out, int Btot) {
  const int tid  = blockIdx.x * blockDim.x + threadIdx.x;
  const int lane = threadIdx.x & 31;                 // wave32
  const bool hi  = (lane & 16) != 0;
  const int wave = tid >> 5;
  const int elem = wave * 16 + (lane & 15);          // 16 rollouts per wave
  const int e    = elem < Btot ? elem : Btot - 1;    // clamp: EXEC stays full

  // ---- loop-invariant: weights in A layout, biases in C layout -------------
  v2f Af1h1 = load_A_stacked(fc1_w, hf1_w, lane);    // fc1 rows 0-3, hf1 4-7
  v2f Afc2  = load_A(fc2_w, 2, 4, 0, lane);
  v2f Ahf2  = load_A(hf2_w, 2, 4, 0, lane);
  v2f Arc1  = load_A(rc1_w, 8, 4, 0, lane);
  v2f Arc2a = load_A(rc2_w, 8, 8, 0, lane);
  v2f Arc2b = load_A(rc2_w, 8, 8, 4, lane);
  v2f Arc3a = load_A(rc3_w, 2, 8, 0, lane);
  v2f Arc3b = load_A(rc3_w, 2, 8, 4, lane);

  v8f C1h1 = bias_C_stacked(fc1_b, hf1_b);
  v8f Cfc2 = bias_C<2>(fc2_b);
  v8f Chf2 = bias_C<2>(hf2_b);
  v8f Crc1 = bias_C<8>(rc1_b);
  v8f Crc2 = bias_C<8>(rc2_b);
  v8f Crc3 = bias_C<2>(rc3_b);

  // s* / s0 per element (lane&15 indexing duplicates across lane halves,
  // which is exactly what the B upper half needs for s*).
  float ss0 = s_star[2 * e + 0];
  float ss1 = s_star[2 * e + 1];
  float sx  = s0[2 * e + 0];
  float sy  = s0[2 * e + 1];
  float err = 0.0f;

#pragma unroll 1
  for (int step = 0; step < ROLL_STEPS; ++step) {
    // x = [s, s*]  -> B (K0=sx, K1=sy, K2=ss0, K3=ss1); s* already duplicated.
    v2f Bx;
    Bx[0] = hi ? ss0 : sx;
    Bx[1] = hi ? ss1 : sy;

    // ---- fused fc1 + hf1: D rows 0-3 = h1, rows 4-7 = h1_hf (no act) -------
    v8f h1 = wmma4(Af1h1, Bx, C1h1);

    // ---- fc2: ah = tanh(fc2(h1)) -------------------------------------------
    v2f Bh1;
    Bh1[0] = upmix(h1[0], h1[2], hi);
    Bh1[1] = upmix(h1[1], h1[3], hi);
    v8f pfc2 = wmma4(Afc2, Bh1, Cfc2);
    float ah0 = tanh_f(pfc2[0]);
    float ah1 = tanh_f(pfc2[1]);

    // ---- hf2 (detached predictor): ah_hat ----------------------------------
    v2f Bg1;
    Bg1[0] = upmix(h1[4], h1[6], hi);
    Bg1[1] = upmix(h1[5], h1[7], hi);
    v8f phf2 = wmma4(Ahf2, Bg1, Chf2);
    float bh0 = tanh_f(phf2[0]);
    float bh1 = tanh_f(phf2[1]);

    // ---- recurrent net on ctx = [s, ah] ------------------------------------
    v2f Bctx;
    Bctx[0] = upmix(sx, ah0, hi);
    Bctx[1] = upmix(sy, ah1, hi);
    v8f r1 = wmma4(Arc1, Bctx, Crc1);
    float t0 = tanh_f(r1[0]), t1 = tanh_f(r1[1]);
    float t2 = tanh_f(r1[2]), t3 = tanh_f(r1[3]);
    float t4 = tanh_f(r1[4]), t5 = tanh_f(r1[5]);
    float t6 = tanh_f(r1[6]), t7 = tanh_f(r1[7]);

    v2f Blo, Bhi2;                                     // 8-wide K split in two
    Blo[0]  = upmix(t0, t2, hi);
    Blo[1]  = upmix(t1, t3, hi);
    Bhi2[0] = upmix(t4, t6, hi);
    Bhi2[1] = upmix(t5, t7, hi);
    v8f r2 = wmma4(Arc2a, Blo, Crc2);
    r2 = wmma4(Arc2b, Bhi2, r2);                       // accumulate K=4..7
    float u0 = tanh_f(r2[0]), u1 = tanh_f(r2[1]);
    float u2 = tanh_f(r2[2]), u3 = tanh_f(r2[3]);
    float u4 = tanh_f(r2[4]), u5 = tanh_f(r2[5]);
    float u6 = tanh_f(r2[6]), u7 = tanh_f(r2[7]);

    v2f B2lo, B2hi;
    B2lo[0] = upmix(u0, u2, hi);
    B2lo[1] = upmix(u1, u3, hi);
    B2hi[0] = upmix(u4, u6, hi);
    B2hi[1] = upmix(u5, u7, hi);
    v8f ar = wmma4(Arc3a, B2lo, Crc3);
    ar = wmma4(Arc3b, B2hi, ar);

    // ---- state + error updates (valid in lanes 0-15) -----------------------
    sx = fmaf(0.1f, ar[0], sx);
    sy = fmaf(0.1f, ar[1], sy);

    float d0 = sx - ss0, d1 = sy - ss1;
    err += d0 * d0 + d1 * d1;
    err += tanh_f(ah0) + tanh_f(ah1);
    err += __builtin_amdgcn_sqrtf(ah0 * ah0 + ah1 * ah1);
    float q0 = ah0 - bh0, q1 = ah1 - bh1;
    err += __builtin_amdgcn_sqrtf(q0 * q0 + q1 * q1);
  }

  if ((lane < 16) && (elem < Btot)) err_out[elem] = err;
}

extern "C" void kernel_launch(void* const* d_in, const int* in_sizes, int n_in,
                              void* d_out, int out_size, void* d_ws,
                              size_t ws_size, hipStream_t stream) {
  const float* s_star = (const float*)d_in[0];
  const float* s0p    = (const float*)d_in[1];
  const float* fc1_w  = (const float*)d_in[2];
  const float* fc1_b  = (const float*)d_in[3];
  const float* fc2_w  = (const float*)d_in[4];
  const float* fc2_b  = (const float*)d_in[5];
  const float* rc1_w  = (const float*)d_in[6];
  const float* rc1_b  = (const float*)d_in[7];
  const float* rc2_w  = (const float*)d_in[8];
  const float* rc2_b  = (const float*)d_in[9];
  const float* rc3_w  = (const float*)d_in[10];
  const float* rc3_b  = (const float*)d_in[11];
  const float* hf1_w  = (const float*)d_in[12];
  const float* hf1_b  = (const float*)d_in[13];
  const float* hf2_w  = (const float*)d_in[14];
  const float* hf2_b  = (const float*)d_in[15];

  const int Btot = in_sizes[0] / 2;                   // s_star is [B,2]
  const long long waves = (Btot + 15) / 16;           // 16 rollouts per wave32
  const long long threads = waves * 32;
  const int block = 256;
  const int grid = (int)((threads + block - 1) / block);

  rollout_wmma_kernel<<<grid, block, 0, stream>>>(
      s_star, s0p, fc1_w, fc1_b, fc2_w, fc2_b, rc1_w, rc1_b, rc2_w, rc2_b,
      rc3_w, rc3_b, hf1_w, hf1_b, hf2_w, hf2_b, (float*)d_out, Btot);
}